// CoordinateMassivePool_17600775979583
// MI455X (gfx1250) — compile-verified
//
#include <hip/hip_runtime.h>

// CoordinateMassivePool — MI455X (gfx1250, wave32)
//
// out[b,:] = normalize(gauss_weights(mu[b],sigma[b])) . params[start:start+256, :]
// HBM-bound: 4096 windows * 256 rows * 512 B = 512 MB of reads with ~no reuse
// (1M-row table, 192 MB L2 can't hold it) -> floor ~22 us @ 23.3 TB/s.
// Per-batch matvec w(1x256).S(256x128) runs through V_WMMA_F32_16X16X4_F32:
// A row M=0 carries the weights, B carries a 4x16 window tile, C row 0 is the
// result. One block per batch; 8 waves; wave n owns columns [16n,16n+16).

typedef __attribute__((ext_vector_type(2))) float v2f;
typedef __attribute__((ext_vector_type(8))) float v8f;

#define TOTAL_N 1000000
#define DIM     128
#define BATCH   4096
#define WIN     256

__global__ __launch_bounds__(256)
void coordpool_wmma_kernel(const float* __restrict__ mu,
                           const float* __restrict__ sigma,
                           const float* __restrict__ P,
                           float* __restrict__ out_agg,
                           int*   __restrict__ out_start)
{
    __shared__ float wbuf[WIN];   // normalized weights (8B-aligned pairs)
    __shared__ float ssum;        // block-wide weight sum

    const int b    = blockIdx.x;
    const int tid  = threadIdx.x;        // 0..255, doubles as window index w
    const int lane = tid & 31;
    const int wave = tid >> 5;           // 0..7 -> 16-column tile

    // ---- window placement (matches reference clip -> int32 truncation) ----
    const float center = mu[b] * (float)(TOTAL_N - 1);
    float startf = center - (float)(WIN / 2);
    startf = fminf(fmaxf(startf, 0.0f), (float)(TOTAL_N - WIN));
    const int start = (int)startf;

    const float sig    = sigma[b] + 1e-6f;
    const float inv2s2 = 1.0f / (2.0f * sig * sig);

    // ---- Gaussian weight for this thread's window row ----
    const float dist = (float)(start + tid) - center;
    const float wgt  = __expf(-(dist * dist) * inv2s2);

    if (tid == 0) ssum = 0.0f;
    __syncthreads();

    // wave32 reduction, one ds_add_f32 per wave into the block sum
    float s = wgt;
    #pragma unroll
    for (int off = 16; off > 0; off >>= 1)
        s += __shfl_xor(s, off, 32);
    if (lane == 0) atomicAdd(&ssum, s);
    __syncthreads();

    wbuf[tid] = wgt / (ssum + 1e-6f);
    __syncthreads();

    // ---- WMMA reduction: C(16x16) += A(16x4) * B(4x16), 64 K-steps ----
    // A f32 16x4 layout: lanes 0-15 -> M=lane, K=0(v0)/K=1(v1);
    //                    lanes 16-31 -> M=lane-16, K=2(v0)/K=3(v1).
    // B assumed to mirror A: v0 = K0|K2 across lane halves, v1 = K1|K3.
    // C: VGPR0 lanes 0-15 = row M=0, cols n0..n0+15 -> our result.
    const int  n0   = wave * 16;
    const int  half = lane >> 4;          // 0 or 1
    const int  col  = lane & 15;
    const bool lead = (col == 0);         // lane carrying A row M=0

    v8f c0 = {};
    v8f c1 = {};
    const float* base = P + (size_t)start * DIM + n0 + col;

    #pragma unroll 4
    for (int k = 0; k < WIN; k += 8) {
        const float* rp0 = base + (size_t)(k     + 2 * half) * DIM;
        const float* rp1 = base + (size_t)(k + 4 + 2 * half) * DIM;

        // stream-ahead prefetch into near caches; covers every row of the
        // stride-2 row pattern (even rows via rp, odd rows via rp+DIM)
        if (k + 40 <= WIN) {
            __builtin_prefetch(rp0 + 32 * DIM, 0, 3);
            __builtin_prefetch(rp0 + 33 * DIM, 0, 3);
            __builtin_prefetch(rp1 + 32 * DIM, 0, 3);
            __builtin_prefetch(rp1 + 33 * DIM, 0, 3);
        }

        // issue ALL fragment loads before either WMMA so they clause and
        // overlap (no full-counter drains between steps)
        float b0x = rp0[0];
        float b0y = rp0[DIM];
        float b1x = rp1[0];
        float b1y = rp1[DIM];

        // unconditional aligned LDS pair loads (ds_load_b64), mask in VALU
        const v2f aw0 = *(const v2f*)&wbuf[k     + 2 * half];
        const v2f aw1 = *(const v2f*)&wbuf[k + 4 + 2 * half];

        v2f A0, A1, B0, B1;
        A0.x = lead ? aw0.x : 0.0f;  A0.y = lead ? aw0.y : 0.0f;
        A1.x = lead ? aw1.x : 0.0f;  A1.y = lead ? aw1.y : 0.0f;
        B0.x = b0x;  B0.y = b0y;
        B1.x = b1x;  B1.y = b1y;

        c0 = __builtin_amdgcn_wmma_f32_16x16x4_f32(false, A0, false, B0,
                                                   (short)0, c0, false, false);
        c1 = __builtin_amdgcn_wmma_f32_16x16x4_f32(false, A1, false, B1,
                                                   (short)0, c1, false, false);
    }

    const v8f c = c0 + c1;

    // row M=0 of C = c[0] on lanes 0..15
    if (half == 0)
        out_agg[(size_t)b * DIM + n0 + col] = c[0];

    if (tid == 0)
        out_start[b] = start;
}

extern "C" void kernel_launch(void* const* d_in, const int* in_sizes, int n_in,
                              void* d_out, int out_size, void* d_ws, size_t ws_size,
                              hipStream_t stream)
{
    const float* mu    = (const float*)d_in[0];
    const float* sigma = (const float*)d_in[1];
    const float* P     = (const float*)d_in[2];

    float* out_agg   = (float*)d_out;                          // 4096*128 f32
    int*   out_start = (int*)(out_agg + (size_t)BATCH * DIM);  // then 4096 i32

    coordpool_wmma_kernel<<<BATCH, 256, 0, stream>>>(mu, sigma, P, out_agg, out_start);
}